// VectorQuantizer_5145370821476
// MI455X (gfx1250) — compile-verified
//
#include <hip/hip_runtime.h>
#include <hip/hip_bf16.h>
#include <math.h>

// VQ-VAE vector quantizer for MI455X (gfx1250, wave32).
// B=64, D=256, T=1024, K=1024, N=B*T=65536.
// Heavy op: [65536 x 256] x [256 x 1024] f32 matmul fused with argmin,
// executed on the matrix pipe via V_WMMA_F32_16X16X4_F32.

#define VQ_D 256
#define VQ_T 1024
#define VQ_K 1024
#define VQ_B 64
#define VQ_N (VQ_B * VQ_T)          // 65536
#define VQ_ELEMS ((size_t)VQ_B * VQ_D * VQ_T)  // 16777216

typedef __attribute__((ext_vector_type(2))) float v2f;
typedef __attribute__((ext_vector_type(8))) float v8f;

// ---------------------------------------------------------------------------
// Kernel 1: per-code squared norms ||e_k||^2  (K=1024)
// ---------------------------------------------------------------------------
__global__ __launch_bounds__(256) void vq_enorm_kernel(
    const float* __restrict__ emb, float* __restrict__ enorm) {
  int k = blockIdx.x * 256 + threadIdx.x;      // grid 4 -> 1024 codes
  const float* row = emb + (size_t)k * VQ_D;
  float s = 0.0f;
  #pragma unroll
  for (int d = 0; d < VQ_D; d += 4) {
    float4 v = *(const float4*)(row + d);
    s += v.x * v.x + v.y * v.y + v.z * v.z + v.w * v.w;
  }
  enorm[k] = s;
}

// ---------------------------------------------------------------------------
// Kernel 2: fused distance-GEMM + argmin using V_WMMA_F32_16X16X4_F32.
// Grid: 512 WGs x 256 threads (8 waves). Wave w of WG g owns rows
// row0 = g*128 + w*16 .. +15 (16 t-values within one batch b).
// A tile (16 rows x 256 depth) held in registers (v2f a[64] = 128 VGPRs).
// B tile (16 codes x 256 depth) staged in LDS, pair-interleaved so each
// WMMA chunk's B operand is one conflict-free ds_load_b64.
// score = 2*(x . e_k) - ||e_k||^2 ; argmax(score) == argmin(dist).
// ---------------------------------------------------------------------------
__global__ __launch_bounds__(256) void vq_argmin_kernel(
    const float* __restrict__ in,     // [B, D, T]
    const float* __restrict__ emb,    // [K, D]
    const float* __restrict__ enorm,  // [K]
    int* __restrict__ idx)            // [N]
{
  __shared__ float ldsB[16 * VQ_D];   // 16 KB

  const int tid  = threadIdx.x;
  const int wave = tid >> 5;
  const int lane = tid & 31;
  const int h    = lane >> 4;         // lane half (0/1)
  const int code = lane & 15;         // N-column within tile / A-row M

  const int row0 = blockIdx.x * 128 + wave * 16;  // 128 rows per WG, one batch
  const int b    = row0 >> 10;
  const int t0   = row0 & (VQ_T - 1);

  // ---- Load A tile into registers (layout for WMMA f32 16x16x4):
  // chunk c, vgpr j (packed in v2f), lane: M = lane&15, K = 4c + j + 2*(lane>>4)
  // flat[row0+M][K] = in[b*D*T + K*T + (t0+M)]
  v2f a[64];
  const float* pin = in + (size_t)b * VQ_D * VQ_T + (size_t)(t0 + code);
  #pragma unroll
  for (int c = 0; c < 64; ++c) {
    const int d0 = 4 * c + 2 * h;
    a[c].x = pin[(size_t)d0 * VQ_T];
    a[c].y = pin[(size_t)(d0 + 1) * VQ_T];
  }

  float best[8];
  int   bidx[8];
  #pragma unroll
  for (int r = 0; r < 8; ++r) { best[r] = -3.402823466e38f; bidx[r] = 0; }

  // LDS read base: B chunk c -> ds_load_b64 at dword ((2c+h)*32 + 2*code)
  const float* ldsRd = ldsB + h * 32 + 2 * code;

  for (int kt = 0; kt < 64; ++kt) {
    const int k0 = kt * 16;
    __syncthreads();
    // Cooperative fill: thread (fc, dc) copies e[k0+fc][dc*16 .. +15] into
    // pair-interleaved LDS: lds[(d>>1)*32 + fc*2 + (d&1)].
    {
      const int fc = tid & 15;
      const int dc = tid >> 4;                     // 0..15
      const float* src = emb + (size_t)(k0 + fc) * VQ_D + dc * 16;
      float vv[16];
      #pragma unroll
      for (int i = 0; i < 16; i += 4) {
        float4 v = *(const float4*)(src + i);
        vv[i] = v.x; vv[i + 1] = v.y; vv[i + 2] = v.z; vv[i + 3] = v.w;
      }
      #pragma unroll
      for (int i = 0; i < 16; ++i) {
        const int d = dc * 16 + i;
        ldsB[(d >> 1) * 32 + fc * 2 + (d & 1)] = vv[i];
      }
    }
    __syncthreads();

    // 64 chained f32 WMMAs: 16x16 tile of x . e^T over depth 256.
    v8f acc = {};
    #pragma unroll
    for (int c = 0; c < 64; ++c) {
      v2f bb = *(const v2f*)(ldsRd + c * 64);
      acc = __builtin_amdgcn_wmma_f32_16x16x4_f32(
          false, a[c], false, bb, (short)0, acc, false, false);
    }

    // Fused argmax update. C layout: acc[r] -> (M = r + 8*h, N = code).
    const float en = enorm[k0 + code];
    #pragma unroll
    for (int r = 0; r < 8; ++r) {
      const float s = 2.0f * acc[r] - en;
      if (s > best[r]) { best[r] = s; bidx[r] = k0 + code; }
    }
  }

  // Reduce across the 16 lanes holding different codes for the same row.
  // Max score, tie-break to lower code index (argmin-first semantics).
  #pragma unroll
  for (int r = 0; r < 8; ++r) {
    float v = best[r];
    int   ix = bidx[r];
    #pragma unroll
    for (int off = 8; off >= 1; off >>= 1) {
      const float ov  = __shfl_xor(v, off, 32);
      const int   oix = __shfl_xor(ix, off, 32);
      if (ov > v || (ov == v && oix < ix)) { v = ov; ix = oix; }
    }
    if (code == 0) idx[row0 + r + 8 * h] = ix;
  }
}

// ---------------------------------------------------------------------------
// Kernel 3: gather-quantize + straight-through output + per-block sq-error.
// One thread per output element, coalesced over [B, D, T].
// ---------------------------------------------------------------------------
__global__ __launch_bounds__(256) void vq_quantize_kernel(
    const float* __restrict__ in, const float* __restrict__ emb,
    const int* __restrict__ idx, float* __restrict__ out,
    float* __restrict__ partial)
{
  __shared__ float red[256];
  const size_t g = (size_t)blockIdx.x * 256 + threadIdx.x;  // grid 65536
  const int t  = (int)(g & (VQ_T - 1));
  const size_t bd = g >> 10;
  const int d  = (int)(bd & (VQ_D - 1));
  const int b  = (int)(bd >> 8);
  const int n  = (b << 10) + t;
  const int k  = idx[n];
  const float x = in[g];
  const float q = emb[(size_t)k * VQ_D + d];
  out[g] = x + (q - x);                 // straight-through forward value
  const float diff = q - x;
  red[threadIdx.x] = diff * diff;
  __syncthreads();
  #pragma unroll
  for (int s = 128; s > 0; s >>= 1) {
    if (threadIdx.x < (unsigned)s) red[threadIdx.x] += red[threadIdx.x + s];
    __syncthreads();
  }
  if (threadIdx.x == 0) partial[blockIdx.x] = red[0];
}

// ---------------------------------------------------------------------------
// Kernel 4: deterministic second-stage loss reduction (65536 partials).
// loss = (1 + commitment_cost) * mse  since e_latent == q_latent in forward.
// ---------------------------------------------------------------------------
__global__ __launch_bounds__(256) void vq_loss_kernel(
    const float* __restrict__ partial, float* __restrict__ loss_out)
{
  __shared__ float red[256];
  float s = 0.0f;
  for (int i = threadIdx.x; i < 65536; i += 256) s += partial[i];
  red[threadIdx.x] = s;
  __syncthreads();
  #pragma unroll
  for (int st = 128; st > 0; st >>= 1) {
    if (threadIdx.x < (unsigned)st) red[threadIdx.x] += red[threadIdx.x + st];
    __syncthreads();
  }
  if (threadIdx.x == 0)
    loss_out[0] = 1.25f * red[0] * (1.0f / (float)VQ_ELEMS);
}

// ---------------------------------------------------------------------------
// Kernels 5-7: code-usage stats.
// ---------------------------------------------------------------------------
__global__ __launch_bounds__(256) void vq_zero_kernel(int* __restrict__ counts) {
  counts[blockIdx.x * 256 + threadIdx.x] = 0;   // grid 4
}

__global__ __launch_bounds__(256) void vq_hist_kernel(
    const int* __restrict__ idx, int* __restrict__ counts)
{
  __shared__ int h[VQ_K];
  for (int i = threadIdx.x; i < VQ_K; i += 256) h[i] = 0;
  __syncthreads();
  const int base = blockIdx.x * 1024;           // grid 64 -> 65536 entries
  for (int i = threadIdx.x; i < 1024; i += 256)
    atomicAdd(&h[idx[base + i]], 1);
  __syncthreads();
  for (int i = threadIdx.x; i < VQ_K; i += 256)
    if (h[i]) atomicAdd(&counts[i], h[i]);
}

__global__ __launch_bounds__(256) void vq_stats_kernel(
    const int* __restrict__ counts, float* __restrict__ ppl_out,
    float* __restrict__ dcr_out)
{
  __shared__ float se[256];
  __shared__ float sa[256];
  float e = 0.0f, act = 0.0f;
  for (int i = threadIdx.x; i < VQ_K; i += 256) {
    const float c = (float)counts[i];
    const float p = c * (1.0f / (float)VQ_N);
    e += p * logf(p + 1e-10f);
    act += (c > 0.0f) ? 1.0f : 0.0f;
  }
  se[threadIdx.x] = e;
  sa[threadIdx.x] = act;
  __syncthreads();
  #pragma unroll
  for (int st = 128; st > 0; st >>= 1) {
    if (threadIdx.x < (unsigned)st) {
      se[threadIdx.x] += se[threadIdx.x + st];
      sa[threadIdx.x] += sa[threadIdx.x + st];
    }
    __syncthreads();
  }
  if (threadIdx.x == 0) {
    ppl_out[0] = expf(-se[0]);
    dcr_out[0] = 1.0f - sa[0] * (1.0f / (float)VQ_K);
  }
}

// ---------------------------------------------------------------------------
// Launch wrapper. d_out layout: [0]=loss, [1..16777216]=out(B,D,T),
// [16777217]=perplexity, [16777218]=dcr.
// ---------------------------------------------------------------------------
extern "C" void kernel_launch(void* const* d_in, const int* in_sizes, int n_in,
                              void* d_out, int out_size, void* d_ws, size_t ws_size,
                              hipStream_t stream) {
  const float* in  = (const float*)d_in[0];   // [64, 256, 1024] f32
  const float* emb = (const float*)d_in[1];   // [1024, 256] f32
  float* out = (float*)d_out;

  char* ws = (char*)d_ws;
  float* enorm   = (float*)ws;                                   //   4 KB
  int*   idx     = (int*)(ws + 4096);                            // 256 KB
  float* partial = (float*)(ws + 4096 + 262144);                 // 256 KB
  int*   counts  = (int*)(ws + 4096 + 262144 + 262144);          //   4 KB

  vq_enorm_kernel<<<4, 256, 0, stream>>>(emb, enorm);
  vq_argmin_kernel<<<512, 256, 0, stream>>>(in, emb, enorm, idx);
  vq_quantize_kernel<<<65536, 256, 0, stream>>>(in, emb, idx, out + 1, partial);
  vq_loss_kernel<<<1, 256, 0, stream>>>(partial, out);
  vq_zero_kernel<<<4, 256, 0, stream>>>(counts);
  vq_hist_kernel<<<64, 256, 0, stream>>>(idx, counts);
  vq_stats_kernel<<<1, 256, 0, stream>>>(counts, out + 1 + VQ_ELEMS,
                                         out + 2 + VQ_ELEMS);
}